// SVAE_89060441850605
// MI455X (gfx1250) — compile-verified
//
#include <hip/hip_runtime.h>
#include <math.h>

// Problem dims (match reference)
#define BB   32
#define TT   256
#define NN   256
#define LL   64
#define RR   16
#define HH   256
#define SS   8
#define BT   (BB*TT)          // 8192
#define OUTD (LL + LL*RR)     // 1088

typedef __attribute__((ext_vector_type(2))) float v2f;
typedef __attribute__((ext_vector_type(8))) float v8f;

#if defined(__HIP_DEVICE_COMPILE__) && defined(__gfx1250__) && \
    __has_builtin(__builtin_amdgcn_wmma_f32_16x16x4_f32)
#define HAVE_WMMA_F32 1
#else
#define HAVE_WMMA_F32 0
#endif

// ---------------------------------------------------------------------------
// Wave-level fp32 WMMA 16x16 tile accumulate: C += A(16xK) * B(Kx16)
// A row-major (lda), B row-major (ldb).  V_WMMA_F32_16X16X4_F32 layouts:
//   A frag: lane r(<16)/half h: a[i] = A[r][k0 + 2h + i]
//   B frag: b[i] = B[k0 + 2h + i][r]
//   C frag: c[v] = C[v + 8h][r]
// ---------------------------------------------------------------------------
__device__ __forceinline__ v8f wmma_tile_f32(const float* __restrict__ A, int lda,
                                             const float* __restrict__ Bm, int ldb,
                                             int K, v8f acc, int lane) {
  const int half = lane >> 4;
  const int r    = lane & 15;
#if HAVE_WMMA_F32
  for (int k0 = 0; k0 < K; k0 += 4) {
    const int ka = k0 + 2*half;
    if ((k0 & 15) == 0) {   // prefetch next K panel (global_prefetch_b8)
      __builtin_prefetch(&A[r*lda + k0 + 16], 0, 0);
      __builtin_prefetch(&Bm[(size_t)(k0 + 16)*ldb + r], 0, 0);
    }
    v2f a, b;
    a.x = A[r*lda + ka];
    a.y = A[r*lda + ka + 1];
    b.x = Bm[(size_t)ka*ldb + r];
    b.y = Bm[(size_t)(ka+1)*ldb + r];
    acc = __builtin_amdgcn_wmma_f32_16x16x4_f32(false, a, false, b, (short)0, acc,
                                                false, false);
  }
#else
  for (int v = 0; v < 8; ++v) {
    const int m = v + 8*half;
    float s = acc[v];
    for (int k = 0; k < K; ++k) s += A[m*lda + k] * Bm[(size_t)k*ldb + r];
    acc[v] = s;
  }
#endif
  return acc;
}

// ---------------------------------------------------------------------------
// Kernel 1: h = tanh(y @ W1 + b1)   (BT x NN) x (NN x HH); 8 waves/block
// ---------------------------------------------------------------------------
__global__ void __launch_bounds__(256) k_enc1(const float* __restrict__ Y,
                                              const float* __restrict__ W1,
                                              const float* __restrict__ b1,
                                              float* __restrict__ Hout) {
  const int wv   = threadIdx.x >> 5;
  const int lane = threadIdx.x & 31;
  const int m0 = blockIdx.x * 16;
  const int n0 = (blockIdx.y * 8 + wv) * 16;
  v8f acc = {};
  acc = wmma_tile_f32(Y + (size_t)m0*NN, NN, W1 + n0, HH, NN, acc, lane);
  const int half = lane >> 4, r = lane & 15;
  const float bias = b1[n0 + r];
  for (int v = 0; v < 8; ++v)
    Hout[(size_t)(m0 + v + 8*half)*HH + n0 + r] = tanhf(acc[v] + bias);
}

// ---------------------------------------------------------------------------
// Kernel 2: out = h @ W2 + b2       (BT x HH) x (HH x OUTD); 8 waves/block
// ---------------------------------------------------------------------------
__global__ void __launch_bounds__(256) k_enc2(const float* __restrict__ Hin,
                                              const float* __restrict__ W2,
                                              const float* __restrict__ b2,
                                              float* __restrict__ Outm) {
  const int wv   = threadIdx.x >> 5;
  const int lane = threadIdx.x & 31;
  const int m0 = blockIdx.x * 16;
  const int n0 = (blockIdx.y * 8 + wv) * 16;
  if (n0 >= OUTD) return;              // OUTD/16 = 68 tiles, gridDim.y = 9
  v8f acc = {};
  acc = wmma_tile_f32(Hin + (size_t)m0*HH, HH, W2 + n0, OUTD, HH, acc, lane);
  const int half = lane >> 4, r = lane & 15;
  const float bias = b2[n0 + r];
  for (int v = 0; v < 8; ++v)
    Outm[(size_t)(m0 + v + 8*half)*OUTD + n0 + r] = acc[v] + bias;
}

// ---------------------------------------------------------------------------
// Kernel 3: J[bt] = K_y K_y^T, K_y = out[bt, 64:].reshape(64,16)
// 8 waves/block, one bt per wave; 4x4 WMMA tiles with K=16.
// ---------------------------------------------------------------------------
__global__ void __launch_bounds__(256) k_J(const float* __restrict__ Outm,
                                           float* __restrict__ Jg) {
  const int wv   = threadIdx.x >> 5;
  const int lane = threadIdx.x & 31;
  const size_t bt = (size_t)blockIdx.x * 8 + wv;
  const float* Km = Outm + bt*OUTD + LL;   // 64x16 row-major
  float* Jb = Jg + bt*(size_t)(LL*LL);
  const int half = lane >> 4, r = lane & 15;
  for (int i0 = 0; i0 < LL; i0 += 16)
    for (int j0 = 0; j0 < LL; j0 += 16) {
      v8f acc = {};
#if HAVE_WMMA_F32
      for (int k0 = 0; k0 < RR; k0 += 4) {
        const int ka = k0 + 2*half;
        v2f a, b;
        a.x = Km[(i0 + r)*RR + ka];
        a.y = Km[(i0 + r)*RR + ka + 1];
        b.x = Km[(j0 + r)*RR + ka];      // B = K^T: B[k][n] = Km[j0+n][k]
        b.y = Km[(j0 + r)*RR + ka + 1];
        acc = __builtin_amdgcn_wmma_f32_16x16x4_f32(false, a, false, b, (short)0,
                                                    acc, false, false);
      }
#else
      for (int v = 0; v < 8; ++v) {
        const int i = i0 + v + 8*half;
        float s = 0.f;
        for (int k = 0; k < RR; ++k) s += Km[i*RR + k] * Km[(j0 + r)*RR + k];
        acc[v] = s;
      }
#endif
      for (int v = 0; v < 8; ++v)
        Jb[(size_t)(i0 + v + 8*half)*LL + j0 + r] = acc[v];
    }
}

// ---------------------------------------------------------------------------
// 64-thread helpers (row i == tid) -- used by the per-(b,t) kernels
// ---------------------------------------------------------------------------
__device__ void chol64_ip(float* S, int ld, int tid) {
  for (int k = 0; k < 64; ++k) {
    if (tid == k) S[k*ld + k] = sqrtf(fmaxf(S[k*ld + k], 1e-30f));
    __syncthreads();
    if (tid > k) S[tid*ld + k] /= S[k*ld + k];
    __syncthreads();
    if (tid > k) {
      const float lik = S[tid*ld + k];
      for (int j = k + 1; j <= tid; ++j) S[tid*ld + j] -= lik * S[j*ld + k];
    }
    __syncthreads();
  }
}

__device__ void tri_solve_vec(const float* Lm, int ld, const float* bv,
                              float* x, float* tmp, int tid) {
  if (tid == 0) {
    for (int i = 0; i < 64; ++i) {
      float s = bv[i];
      for (int j = 0; j < i; ++j) s -= Lm[i*ld + j] * tmp[j];
      tmp[i] = s / Lm[i*ld + i];
    }
    for (int i = 63; i >= 0; --i) {
      float s = tmp[i];
      for (int j = i + 1; j < 64; ++j) s -= Lm[j*ld + i] * x[j];
      x[i] = s / Lm[i*ld + i];
    }
  }
  __syncthreads();
}

// ---------------------------------------------------------------------------
// 256-thread helpers: row i = tid>>2, quarter q = tid&3 (inner j loop split with stride 4, offset q)
// ---------------------------------------------------------------------------
__device__ void chol64_256(float* S, int ld, int i, int q) {
  for (int k = 0; k < 64; ++k) {
    if (i == k && q == 0) S[k*ld + k] = sqrtf(fmaxf(S[k*ld + k], 1e-30f));
    __syncthreads();
    if (i > k && q == 0) S[i*ld + k] /= S[k*ld + k];
    __syncthreads();
    if (i > k) {
      const float lik = S[i*ld + k];
      for (int j = k + 1 + q; j <= i; j += 4) S[i*ld + j] -= lik * S[j*ld + k];
    }
    __syncthreads();
  }
}

// Solve (L L^T) X = X in place; 4-lane team per column c, intra-wave
// __shfl_xor reduction over the team (lanes 4c..4c+3 are in one wave32;
// same-wave LDS ops are in-order so lane-0 writes are visible to the team).
__device__ void chol_solve_mat_ip4(const float* Lm, int ld, float* X, int ldx,
                                   int c, int q) {
  for (int i = 0; i < 64; ++i) {
    float s = 0.f;
    for (int j = q; j < i; j += 4) s += Lm[i*ld + j] * X[j*ldx + c];
    s += __shfl_xor(s, 1, 4);
    s += __shfl_xor(s, 2, 4);
    const float val = (X[i*ldx + c] - s) / Lm[i*ld + i];
    if (q == 0) X[i*ldx + c] = val;
  }
  for (int i = 63; i >= 0; --i) {
    float s = 0.f;
    for (int j = i + 1 + q; j < 64; j += 4) s += Lm[j*ld + i] * X[j*ldx + c];
    s += __shfl_xor(s, 1, 4);
    s += __shfl_xor(s, 2, 4);
    const float val = (X[i*ldx + c] - s) / Lm[i*ld + i];
    if (q == 0) X[i*ldx + c] = val;
  }
  __syncthreads();
}

// ---------------------------------------------------------------------------
// Kernel 4: Kalman information filter. 256 threads per batch b.
// h_p = J_p m_p and m_f = P_f_raw (h_p + k_t): identical math to cho_solve
// since J_p = P_p^{-1}, P_f_raw = J_f^{-1} (removes serial vector solves).
// ---------------------------------------------------------------------------
__global__ void __launch_bounds__(256) k_filter(
    const float* __restrict__ Outm, const float* __restrict__ Jg,
    const float* __restrict__ Fm, const float* __restrict__ logQ,
    const float* __restrict__ logQ0, const float* __restrict__ m0v,
    float* __restrict__ mf, float* __restrict__ mp,
    float* __restrict__ Pf, float* __restrict__ Pp) {
  __shared__ float sA[64*65];
  __shared__ float sB[64*64];
  __shared__ float sC[64*64];
  __shared__ float v_prev[64], v_mp[64], v_hp[64], v_rhs[64];
  const int b   = blockIdx.x;
  const int tid = threadIdx.x;
  const int i   = tid >> 2;
  const int q   = tid & 3;
  const int j0  = q * 16;
  const float Qd  = log1pf(expf(logQ[i]));
  const float Q0d = log1pf(expf(logQ0[i]));

  for (int t = 0; t < TT; ++t) {
    const size_t bt = (size_t)b*TT + t;
    if (t == 0) {
      for (int j = j0; j < j0 + 16; ++j) sA[i*65 + j] = (j == i) ? Q0d : 0.0f;
      if (q == 0) v_mp[i] = m0v[i];
    } else {
      if (q == 0) {
        float s = 0.f;
        for (int j = 0; j < 64; ++j) s += Fm[i*64 + j] * v_prev[j];
        v_mp[i] = s;
      }
      for (int j = j0; j < j0 + 16; ++j) {        // sC = F @ P_prev(sB)
        float a = 0.f;
        for (int k = 0; k < 64; ++k) a += Fm[i*64 + k] * sB[k*64 + j];
        sC[i*64 + j] = a;
      }
      __syncthreads();
      for (int j = j0; j < j0 + 16; ++j) {        // P_p = sC F^T + diag(Qd)
        float a = 0.f;
        for (int k = 0; k < 64; ++k) a += sC[i*64 + k] * Fm[j*64 + k];
        sA[i*65 + j] = a + ((j == i) ? Qd : 0.0f);
      }
    }
    __syncthreads();
    if (q == 0) mp[bt*64 + i] = v_mp[i];
    for (int j = j0; j < j0 + 16; ++j) Pp[bt*4096 + (size_t)i*64 + j] = sA[i*65 + j];
    __syncthreads();

    chol64_256(sA, 65, i, q);                     // Pp_chol
    for (int j = j0; j < j0 + 16; ++j) sC[i*64 + j] = (j == i) ? 1.0f : 0.0f;
    __syncthreads();
    chol_solve_mat_ip4(sA, 65, sC, 64, i, q);     // J_p = P_p^{-1}
    if (q == 0) {                                 // h_p = J_p m_p
      float s = 0.f;
      for (int k = 0; k < 64; ++k) s += sC[i*64 + k] * v_mp[k];
      v_hp[i] = s;
    }
    {                                             // J_f sym into sA
      const float* Jt = Jg + bt*4096;
      for (int j = j0; j < j0 + 16; ++j)
        sA[i*65 + j] = 0.5f*(sC[i*64 + j] + sC[j*64 + i])
                     + 0.5f*(Jt[(size_t)i*64 + j] + Jt[(size_t)j*64 + i]);
    }
    __syncthreads();
    chol64_256(sA, 65, i, q);                     // Jf_chol
    if (q == 0) v_rhs[i] = v_hp[i] + Outm[bt*OUTD + i];
    for (int j = j0; j < j0 + 16; ++j) sC[i*64 + j] = (j == i) ? 1.0f : 0.0f;
    __syncthreads();
    chol_solve_mat_ip4(sA, 65, sC, 64, i, q);     // P_f raw = J_f^{-1}
    if (q == 0) {                                 // m_f = P_f_raw rhs
      float s = 0.f;
      for (int k = 0; k < 64; ++k) s += sC[i*64 + k] * v_rhs[k];
      v_prev[i] = s;
    }
    for (int j = j0; j < j0 + 16; ++j)
      sB[i*64 + j] = 0.5f*(sC[i*64 + j] + sC[j*64 + i]);
    __syncthreads();
    if (q == 0) mf[bt*64 + i] = v_prev[i];
    for (int j = j0; j < j0 + 16; ++j) Pf[bt*4096 + (size_t)i*64 + j] = sB[i*64 + j];
    __syncthreads();
  }
}

// ---------------------------------------------------------------------------
// Kernel 5: RTS smoother (backward scan). 256 threads per batch.
// ---------------------------------------------------------------------------
__global__ void __launch_bounds__(256) k_smoother(
    const float* __restrict__ Fm,
    const float* __restrict__ mf, const float* __restrict__ mp,
    const float* __restrict__ Pf, const float* __restrict__ Pp,
    float* __restrict__ ms, float* __restrict__ Ps) {
  __shared__ float sA[64*65];
  __shared__ float sB[64*64];   // carry P_s
  __shared__ float sC[64*64];   // F P_f -> X (G = X^T)
  __shared__ float v_ms[64], v_d[64];
  const int b   = blockIdx.x;
  const int tid = threadIdx.x;
  const int i   = tid >> 2;
  const int q   = tid & 3;
  const int j0  = q * 16;
  {
    const size_t bt = (size_t)b*TT + (TT - 1);
    if (q == 0) { v_ms[i] = mf[bt*64 + i]; ms[bt*64 + i] = v_ms[i]; }
    for (int j = j0; j < j0 + 16; ++j) {
      const float v = Pf[bt*4096 + (size_t)i*64 + j];
      sB[i*64 + j] = v;
      Ps[bt*4096 + (size_t)i*64 + j] = v;
    }
  }
  __syncthreads();
  for (int t = TT - 2; t >= 0; --t) {
    const size_t btt = (size_t)b*TT + t;
    const size_t btn = btt + 1;
    for (int j = j0; j < j0 + 16; ++j)
      sA[i*65 + j] = Pp[btn*4096 + (size_t)i*64 + j];
    __syncthreads();
    chol64_256(sA, 65, i, q);
    for (int j = j0; j < j0 + 16; ++j) {          // sC = F @ P_f_t
      float s = 0.f;
      for (int k = 0; k < 64; ++k) s += Fm[i*64 + k] * Pf[btt*4096 + (size_t)k*64 + j];
      sC[i*64 + j] = s;
    }
    __syncthreads();
    chol_solve_mat_ip4(sA, 65, sC, 64, i, q);     // X; G[i][k] = sC[k*64+i]
    if (q == 0) v_d[i] = v_ms[i] - mp[btn*64 + i];
    __syncthreads();
    if (q == 0) {                                 // m_s = m_f + G v_d
      float s = mf[btt*64 + i];
      for (int k = 0; k < 64; ++k) s += sC[k*64 + i] * v_d[k];
      v_ms[i] = s;
    }
    for (int j = j0; j < j0 + 16; ++j) {          // T2 = G (P_s_n - P_p_n)
      float s = 0.f;
      for (int k = 0; k < 64; ++k)
        s += sC[k*64 + i] * (sB[k*64 + j] - Pp[btn*4096 + (size_t)k*64 + j]);
      sA[i*65 + j] = s;
    }
    __syncthreads();
    for (int j = j0; j < j0 + 16; ++j) {          // P_s = P_f + T2 G^T
      float s = Pf[btt*4096 + (size_t)i*64 + j];
      for (int k = 0; k < 64; ++k) s += sA[i*65 + k] * sC[k*64 + j];
      sB[i*64 + j] = s;
    }
    __syncthreads();
    for (int j = j0; j < j0 + 16 && j < i; ++j) { // symmetrize (pair @ max idx)
      const float m2 = 0.5f*(sB[i*64 + j] + sB[j*64 + i]);
      sB[i*64 + j] = m2; sB[j*64 + i] = m2;
    }
    __syncthreads();
    if (q == 0) ms[btt*64 + i] = v_ms[i];
    for (int j = j0; j < j0 + 16; ++j)
      Ps[btt*4096 + (size_t)i*64 + j] = sB[i*64 + j];
    __syncthreads();
  }
}

// ---------------------------------------------------------------------------
// Kernel 6: per-(b,t) KL term (generic alt_kl_step_t; step 0 is the same math
// since m_p[:,0]=m_0 and P_p[:,0]=diag(Q0) were stored by the filter).
// ---------------------------------------------------------------------------
__global__ void __launch_bounds__(64) k_kl(
    const float* __restrict__ Outm,
    const float* __restrict__ ms, const float* __restrict__ mfg,
    const float* __restrict__ mpg,
    const float* __restrict__ Ps, const float* __restrict__ Pp,
    float* __restrict__ klv) {
  const size_t bt = blockIdx.x;
  const int tid = threadIdx.x;
  __shared__ float sA[64*65];
  __shared__ float sAm[64*RR];
  __shared__ float sPA[64*RR];
  __shared__ float sM[16*16];
  __shared__ float red[64];
  __shared__ float v_in[64], v_out[64], v_tmp[64];
  __shared__ float terms[6];

  for (int r = 0; r < RR; ++r) sAm[tid*RR + r] = Outm[bt*OUTD + LL + tid*RR + r];
  for (int j = 0; j < 64; ++j) sA[tid*65 + j] = Pp[bt*4096 + (size_t)tid*64 + j];
  __syncthreads();
  chol64_ip(sA, 65, tid);

  v_in[tid] = mfg[bt*64 + tid];                 // terms[0] = m_f^T Pp^-1 m_f
  __syncthreads();
  tri_solve_vec(sA, 65, v_in, v_out, v_tmp, tid);
  red[tid] = v_in[tid] * v_out[tid];
  __syncthreads();
  if (tid == 0) { float s = 0; for (int k = 0; k < 64; ++k) s += red[k]; terms[0] = s; }
  __syncthreads();
  v_in[tid] = mpg[bt*64 + tid];                 // terms[1] = m_p^T Pp^-1 m_p
  __syncthreads();
  tri_solve_vec(sA, 65, v_in, v_out, v_tmp, tid);
  red[tid] = v_in[tid] * v_out[tid];
  __syncthreads();
  if (tid == 0) { float s = 0; for (int k = 0; k < 64; ++k) s += red[k]; terms[1] = s; }
  __syncthreads();
  for (int r = 0; r < RR; ++r) {                // terms[2] = tr(A^T P_s A)
    float s = 0.f;
    for (int k = 0; k < 64; ++k) s += Ps[bt*4096 + (size_t)tid*64 + k] * sAm[k*RR + r];
    sPA[tid*RR + r] = s;
  }
  { float s = 0.f; for (int r = 0; r < RR; ++r) s += sAm[tid*RR + r]*sPA[tid*RR + r]; red[tid] = s; }
  __syncthreads();
  if (tid == 0) { float s = 0; for (int k = 0; k < 64; ++k) s += red[k]; terms[2] = s; }
  __syncthreads();
  v_in[tid] = ms[bt*64 + tid];                  // terms[3] = a . m_s
  red[tid] = Outm[bt*OUTD + tid] * v_in[tid];
  __syncthreads();
  if (tid == 0) { float s = 0; for (int k = 0; k < 64; ++k) s += red[k]; terms[3] = s; }
  __syncthreads();
  if (tid < RR) {                               // terms[4] = |A^T m_s|^2
    float s = 0.f; for (int k = 0; k < 64; ++k) s += sAm[k*RR + tid]*v_in[k]; red[tid] = s*s;
  }
  __syncthreads();
  if (tid == 0) { float s = 0; for (int r = 0; r < RR; ++r) s += red[r]; terms[4] = s; }
  __syncthreads();
  v_in[tid] = mfg[bt*64 + tid];                 // terms[5] = |A^T m_f|^2
  __syncthreads();
  if (tid < RR) {
    float s = 0.f; for (int k = 0; k < 64; ++k) s += sAm[k*RR + tid]*v_in[k]; red[tid] = s*s;
  }
  __syncthreads();
  if (tid == 0) { float s = 0; for (int r = 0; r < RR; ++r) s += red[r]; terms[5] = s; }
  __syncthreads();
  for (int r = 0; r < RR; ++r) {                // M = I + A^T P_p A
    float s = 0.f;
    for (int k = 0; k < 64; ++k) s += Pp[bt*4096 + (size_t)tid*64 + k] * sAm[k*RR + r];
    sPA[tid*RR + r] = s;
  }
  __syncthreads();
  for (int qq = 0; qq < 4; ++qq) {
    const int idx = tid*4 + qq, r1 = idx >> 4, r2 = idx & 15;
    float s = (r1 == r2) ? 1.0f : 0.0f;
    for (int k = 0; k < 64; ++k) s += sAm[k*RR + r1] * sPA[k*RR + r2];
    sM[r1*16 + r2] = s;
  }
  __syncthreads();
  if (tid == 0) {
    float ldh = 0.f;
    for (int k = 0; k < 16; ++k) {              // 16x16 cholesky (serial)
      float d = sM[k*16 + k];
      for (int j = 0; j < k; ++j) d -= sM[k*16 + j]*sM[k*16 + j];
      d = sqrtf(fmaxf(d, 1e-30f));
      sM[k*16 + k] = d; ldh += logf(d);
      for (int r = k + 1; r < 16; ++r) {
        float s = sM[r*16 + k];
        for (int j = 0; j < k; ++j) s -= sM[r*16 + j]*sM[k*16 + j];
        sM[r*16 + k] = s / d;
      }
    }
    const float logdet = 2.0f*ldh;
    const float inner_p = terms[3] - 0.5f*terms[4] - 0.5f*terms[2];
    const float dlogZ   = 0.5f*(terms[0] - terms[1] + terms[5] - logdet);
    klv[bt] = inner_p - dlogZ;
  }
}

// ---------------------------------------------------------------------------
// Kernel 7: sampling + Poisson readout ELL. One 64-thread block per (b,t).
// ---------------------------------------------------------------------------
__global__ void __launch_bounds__(64) k_sample(
    const float* __restrict__ ms, const float* __restrict__ Ps,
    const float* __restrict__ eps, const float* __restrict__ Cm,
    const float* __restrict__ b_read, const float* __restrict__ Y,
    float* __restrict__ ellv) {
  const size_t bt = blockIdx.x;
  const int tid = threadIdx.x;
  __shared__ float sA[64*65];
  __shared__ float zv[64];
  __shared__ float red[64];
  for (int j = 0; j < 64; ++j)
    sA[tid*65 + j] = Ps[bt*4096 + (size_t)tid*64 + j] + ((j == tid) ? 1e-5f : 0.0f);
  __syncthreads();
  chol64_ip(sA, 65, tid);
  const float m_i = ms[bt*64 + tid];
  float acc = 0.0f;
  for (int s = 0; s < SS; ++s) {
    const float* e = eps + ((size_t)s*BT + bt)*64;
    float z = m_i;
    for (int j = 0; j <= tid; ++j) z += sA[tid*65 + j] * e[j];
    zv[tid] = z;
    __syncthreads();
    for (int qn = 0; qn < 4; ++qn) {
      const int n = tid + 64*qn;
      float lr = b_read[n];
      for (int k = 0; k < 64; ++k) lr += zv[k] * Cm[(size_t)n*64 + k];
      acc += Y[bt*NN + n]*lr - expf(lr);
    }
    __syncthreads();
  }
  red[tid] = acc;
  __syncthreads();
  if (tid == 0) {
    float s = 0.f; for (int k = 0; k < 64; ++k) s += red[k];
    ellv[bt] = s / (float)SS;
  }
}

// ---------------------------------------------------------------------------
// Kernel 8: loss = mean_b sum_t (kl - ell) = sum(kl-ell)/B
// ---------------------------------------------------------------------------
__global__ void __launch_bounds__(256) k_loss(const float* __restrict__ klv,
                                              const float* __restrict__ ellv,
                                              float* __restrict__ out) {
  __shared__ float red[256];
  const int tid = threadIdx.x;
  float s = 0.f;
  for (int k = tid; k < BT; k += 256) s += klv[k] - ellv[k];
  red[tid] = s;
  __syncthreads();
  if (tid == 0) {
    float t = 0.f; for (int k = 0; k < 256; ++k) t += red[k];
    out[0] = t / (float)BB;
  }
}

// ---------------------------------------------------------------------------
extern "C" void kernel_launch(void* const* d_in, const int* in_sizes, int n_in,
                              void* d_out, int out_size, void* d_ws, size_t ws_size,
                              hipStream_t stream) {
  (void)in_sizes; (void)n_in; (void)out_size; (void)ws_size;
  const float* y      = (const float*)d_in[0];
  const float* eps    = (const float*)d_in[1];
  const float* W1     = (const float*)d_in[2];
  const float* b1     = (const float*)d_in[3];
  const float* W2     = (const float*)d_in[4];
  const float* b2     = (const float*)d_in[5];
  const float* Fm     = (const float*)d_in[6];
  const float* logQ   = (const float*)d_in[7];
  const float* logQ0  = (const float*)d_in[8];
  const float* m0v    = (const float*)d_in[9];
  const float* Cm     = (const float*)d_in[10];
  const float* b_read = (const float*)d_in[11];

  float* ws = (float*)d_ws;
  size_t off = 0;
  float* h    = ws + off; off += (size_t)BT*HH;
  float* outm = ws + off; off += (size_t)BT*OUTD;
  float* Jg   = ws + off; off += (size_t)BT*LL*LL;
  float* mf   = ws + off; off += (size_t)BT*LL;
  float* mp   = ws + off; off += (size_t)BT*LL;
  float* msv  = ws + off; off += (size_t)BT*LL;
  float* Pf   = ws + off; off += (size_t)BT*LL*LL;
  float* Pp   = ws + off; off += (size_t)BT*LL*LL;
  float* Ps   = ws + off; off += (size_t)BT*LL*LL;
  float* klv  = ws + off; off += (size_t)BT;
  float* ellv = ws + off; off += (size_t)BT;

  k_enc1<<<dim3(BT/16, (HH/16)/8), 256, 0, stream>>>(y, W1, b1, h);
  k_enc2<<<dim3(BT/16, (OUTD/16 + 7)/8), 256, 0, stream>>>(h, W2, b2, outm);
  k_J<<<BT/8, 256, 0, stream>>>(outm, Jg);
  k_filter<<<BB, 256, 0, stream>>>(outm, Jg, Fm, logQ, logQ0, m0v, mf, mp, Pf, Pp);
  k_smoother<<<BB, 256, 0, stream>>>(Fm, mf, mp, Pf, Pp, msv, Ps);
  k_kl<<<BT, 64, 0, stream>>>(outm, msv, mf, mp, Ps, Pp, klv);
  k_sample<<<BT, 64, 0, stream>>>(msv, Ps, eps, Cm, b_read, y, ellv);
  k_loss<<<1, 256, 0, stream>>>(klv, ellv, (float*)d_out);
}